// GuidedMoEBasic_20615843020914
// MI455X (gfx1250) — compile-verified
//
#include <hip/hip_runtime.h>
#include <hip/hip_bf16.h>

// Problem constants (match reference)
#define BATCH   8
#define DLEN    128
#define HDIM    768
#define NEMO    7
#define UT      776          // H + NE + 1
#define PAIRDIM 1552         // 2*UT
#define NPB     8256         // D*(D+1)/2
#define NROWS   66048        // BATCH * NPB
#define NEXP    2
#define HID     256

#define MROWS   32           // pair rows per workgroup (2 M-tiles)
#define KCH     388          // K per staged chunk; 4 chunks = 1552
#define KCH16   97           // 16-byte units per row chunk (388*4/16)
#define LDSA_STRIDE 396      // floats; 396*4 % 16 == 0 (B128) and bank-conflict-free
#define LDSH_STRIDE 514      // floats; 8*514 % 64 != 0 -> conflict-free h store

typedef __attribute__((ext_vector_type(2))) float v2f;
typedef __attribute__((ext_vector_type(8))) float v8f;

// Map flat pair index (tril row-major: for end, for t<=end) to (b, t, end).
__device__ __forceinline__ void decode_pair(int n, int& b, int& t, int& e) {
    b = n / NPB;
    int q = n - b * NPB;
    int end = (int)((sqrtf(8.0f * (float)q + 1.0f) - 1.0f) * 0.5f);
    while ((end + 1) * (end + 2) / 2 <= q) ++end;   // fixup for fp rounding
    while (end * (end + 1) / 2 > q) --end;
    t = q - end * (end + 1) / 2;
    e = end;
}

// ---------------------------------------------------------------------------
// Kernel 1: emotion head + build utter = [pooled(768) | emo(7) | speaker(1)]
// ---------------------------------------------------------------------------
__global__ __launch_bounds__(256) void build_utter_kernel(
    const float* __restrict__ pooled, const float* __restrict__ spk,
    const float* __restrict__ We, const float* __restrict__ be,
    float* __restrict__ utter, float* __restrict__ emo_out)
{
    const int row = blockIdx.x;           // 0..1023
    const int tid = threadIdx.x;
    const float* src = pooled + (size_t)row * HDIM;
    float* dst = utter + (size_t)row * UT;

    for (int j = tid; j < HDIM; j += 256) dst[j] = src[j];

    if (tid < NEMO) {
        float acc = be[tid];
        for (int k = 0; k < HDIM; ++k) acc += src[k] * We[k * NEMO + tid];
        emo_out[row * NEMO + tid] = acc;
        dst[HDIM + tid] = acc;
    }
    if (tid == NEMO) dst[HDIM + NEMO] = spk[row];
}

// ---------------------------------------------------------------------------
// Kernel 2: gating[n][ex] = concat(utter[t], utter[e]) . Wg[:,ex] + bg[ex]
// ---------------------------------------------------------------------------
__global__ __launch_bounds__(256) void gating_kernel(
    const float* __restrict__ utter, const float* __restrict__ Wg,
    const float* __restrict__ bg, float* __restrict__ gate)
{
    const int idx = blockIdx.x * blockDim.x + threadIdx.x;
    if (idx >= NROWS * NEXP) return;
    const int n = idx >> 1;
    const int ex = idx & 1;
    int b, t, e;
    decode_pair(n, b, t, e);
    const float* ut = utter + (size_t)(b * DLEN + t) * UT;
    const float* ue = utter + (size_t)(b * DLEN + e) * UT;
    float acc = bg[ex];
    for (int j = 0; j < UT; ++j) acc += ut[j] * Wg[j * NEXP + ex];
    for (int j = 0; j < UT; ++j) acc += ue[j] * Wg[(UT + j) * NEXP + ex];
    gate[idx] = acc;
}

// ---------------------------------------------------------------------------
// Kernel 3: fused pair-GEMM (fp32 WMMA) + second layer + gated sum.
// Workgroup = 32 pair rows x 512 cols (2 experts x 256).
// Wave w: expert w/4, N-tiles (w%4)*4..+3, BOTH M-tiles (B-frag reuse x2).
// A staged to LDS in four 388-float K-chunks via async global->LDS B128.
// ---------------------------------------------------------------------------
__global__ __launch_bounds__(256) void moe_wmma_kernel(
    const float* __restrict__ utter,
    const float* __restrict__ W1p, const float* __restrict__ b1p,
    const float* __restrict__ W2p, const float* __restrict__ b2p,
    const float* __restrict__ gate, float* __restrict__ cause)
{
    __shared__ __align__(16) float lds[MROWS * LDSA_STRIDE]; // 50,688 B; reused for h
    __shared__ int roff[MROWS][2];     // per-row utter offsets: [0]=t-row, [1]=e-row

    const int tid  = threadIdx.x;
    const int lane = tid & 31;
    const int wave = tid >> 5;
    const int row0 = blockIdx.x * MROWS;

    if (tid < MROWS) {
        int b, t, e;
        decode_pair(row0 + tid, b, t, e);
        roff[tid][0] = (b * DLEN + t) * UT;
        roff[tid][1] = (b * DLEN + e) * UT;
    }

    const int ex    = wave >> 2;              // expert 0/1
    const int wq    = wave & 3;               // quad of N-tiles within expert
    const int m     = lane & 15;              // A row held by this lane
    const int khalf = (lane >> 4) * 2;        // lanes 16-31 hold k+2,k+3
    const int cloc  = lane & 15;              // column within tile
    const int cbase = wq * 64 + cloc;         // within-expert column, tile i=0
    const float* W1e = W1p + (size_t)ex * (PAIRDIM * HID);

    v8f acc[2][4];
#pragma unroll
    for (int mt = 0; mt < 2; ++mt)
#pragma unroll
        for (int i = 0; i < 4; ++i)
            acc[mt][i] = (v8f){0.f, 0.f, 0.f, 0.f, 0.f, 0.f, 0.f, 0.f};

    __syncthreads();                          // roff visible

    for (int q = 0; q < 4; ++q) {
        if (q) __syncthreads();               // previous chunk's K-loop done
        // ---- async stage: 32 rows x 388 floats (97 x 16B units per row) ----
        const int halfsel = q >> 1;           // 0: t-row half, 1: e-row half
        const int koff    = (q & 1) * KCH;    // float offset within the 776 half
        for (int f = tid; f < MROWS * KCH16; f += 256) {
            const int r = f / KCH16;
            const int j = f - r * KCH16;      // 16B unit index
            const float* gsrc = utter + roff[r][halfsel] + koff + j * 4;
            const unsigned lofs =
                (unsigned)(unsigned long long)(lds + r * LDSA_STRIDE + j * 4);
            const unsigned long long ga = (unsigned long long)gsrc;
            asm volatile("global_load_async_to_lds_b128 %0, %1, off"
                         :: "v"(lofs), "v"(ga)
                         : "memory");
        }
        asm volatile("s_wait_asynccnt 0x0" ::: "memory");
        __syncthreads();

        // ---- WMMA K-loop over this chunk ----
        const float* Wh = W1e + (size_t)(q * KCH) * HID;
        for (int kk = 0; kk < KCH; kk += 4) {
            const int k0 = kk + khalf;
            v2f a0, a1;
            a0.x = lds[m * LDSA_STRIDE + k0];
            a0.y = lds[m * LDSA_STRIDE + k0 + 1];
            a1.x = lds[(16 + m) * LDSA_STRIDE + k0];
            a1.y = lds[(16 + m) * LDSA_STRIDE + k0 + 1];
            const float* bp = Wh + (size_t)k0 * HID + cbase;
#pragma unroll
            for (int i = 0; i < 4; ++i) {
                v2f bf;
                bf.x = bp[i * 16];
                bf.y = bp[HID + i * 16];
                acc[0][i] = __builtin_amdgcn_wmma_f32_16x16x4_f32(
                    false, a0, false, bf, (short)0, acc[0][i], false, false);
                acc[1][i] = __builtin_amdgcn_wmma_f32_16x16x4_f32(
                    false, a1, false, bf, (short)0, acc[1][i], false, false);
            }
        }
    }

    // ---- epilogue: per M-tile, h -> LDS, then h@W2 + b2, gated sum ----
    float* hbuf = lds;
    const int mb = (lane >> 4) * 8;           // C/D: lanes 16-31 hold rows 8..15
    for (int mt = 0; mt < 2; ++mt) {
        __syncthreads();                      // lds free (A reads / prior pass done)
#pragma unroll
        for (int i = 0; i < 4; ++i) {
            const int c = wq * 64 + i * 16 + cloc;     // within-expert column
            const float bias = b1p[ex * HID + c];
            const int colg = ex * HID + c;             // 0..511
#pragma unroll
            for (int v = 0; v < 8; ++v)
                hbuf[(mb + v) * LDSH_STRIDE + colg] = acc[mt][i][v] + bias;
        }
        __syncthreads();
        if (tid < 32) {
            const int mr = tid >> 1;
            const int c  = tid & 1;
            const int n  = row0 + mt * 16 + mr;
            float s0 = b2p[c];                 // b2[0][c]
            float s1 = b2p[2 + c];             // b2[1][c]
            const float* hr = hbuf + mr * LDSH_STRIDE;
            for (int j = 0; j < HID; ++j) {
                s0 += hr[j]       * W2p[j * 2 + c];
                s1 += hr[HID + j] * W2p[(HID + j) * 2 + c];
            }
            cause[n * 2 + c] = gate[n * 2 + 0] * s0 + gate[n * 2 + 1] * s1;
        }
    }
}

// ---------------------------------------------------------------------------
extern "C" void kernel_launch(void* const* d_in, const int* in_sizes, int n_in,
                              void* d_out, int out_size, void* d_ws, size_t ws_size,
                              hipStream_t stream) {
    (void)in_sizes; (void)n_in; (void)out_size; (void)ws_size;
    const float* pooled = (const float*)d_in[0];
    const float* spk    = (const float*)d_in[1];
    const float* We     = (const float*)d_in[2];
    const float* be     = (const float*)d_in[3];
    const float* Wg     = (const float*)d_in[4];
    const float* bg     = (const float*)d_in[5];
    const float* W1     = (const float*)d_in[6];
    const float* b1     = (const float*)d_in[7];
    const float* W2     = (const float*)d_in[8];
    const float* b2     = (const float*)d_in[9];

    float* out   = (float*)d_out;
    float* emo   = out;                                   // [1024*7]
    float* cause = out + BATCH * DLEN * NEMO;             // [66048*2]

    float* utter = (float*)d_ws;                          // [1024*776]
    float* gate  = utter + (size_t)BATCH * DLEN * UT;     // [66048*2]

    build_utter_kernel<<<BATCH * DLEN, 256, 0, stream>>>(pooled, spk, We, be,
                                                         utter, emo);
    gating_kernel<<<(NROWS * NEXP + 255) / 256, 256, 0, stream>>>(utter, Wg, bg,
                                                                  gate);
    moe_wmma_kernel<<<NROWS / MROWS, 256, 0, stream>>>(utter, W1, b1, W2, b2,
                                                       gate, cause);
}